// Net_72653666779710
// MI455X (gfx1250) — compile-verified
//
#include <hip/hip_runtime.h>
#include <hip/hip_bf16.h>
#include <math.h>

typedef __attribute__((ext_vector_type(2))) float v2f;
typedef __attribute__((ext_vector_type(8))) float v8f;

#define GCN_ALPHA 0.1f

__device__ __forceinline__ void atomic_add_f32(float* addr, float v) {
  // native global_atomic_add_f32 on CDNA5
  unsafeAtomicAdd(addr, v);
}

// ---------------- degree / normalization ----------------
__global__ void k_init_deg(float* deg, int N) {
  int i = blockIdx.x * blockDim.x + threadIdx.x;
  if (i < N) deg[i] = 1.0f;  // self-loop contribution
}

__global__ void k_deg_accum(const int* __restrict__ col, float* __restrict__ deg, int E) {
  int e = blockIdx.x * blockDim.x + threadIdx.x;
  if (e < E) atomic_add_f32(&deg[col[e]], 1.0f);
}

__global__ void k_make_dis(float* deg, int N) {
  int i = blockIdx.x * blockDim.x + threadIdx.x;
  if (i < N) {
    float d = deg[i];
    deg[i] = (d > 0.f) ? rsqrtf(d) : 0.f;
  }
}

// ---------------- sparse propagate ----------------
// dst[i,:] = dis[i]^2 * src[i,:]  (self loop; doubles as zero-init). float4 path.
template<int F>
__global__ void k_self_init4(const float* __restrict__ src, const float* __restrict__ dis,
                             float* __restrict__ dst, int N) {
  static_assert(F % 4 == 0, "F must be multiple of 4");
  constexpr int G = F / 4;
  unsigned int i = blockIdx.x * blockDim.x + threadIdx.x;   // float4 index
  if (i < (unsigned int)N * G) {
    unsigned int node = i / G;
    float d = dis[node];
    d = d * d;
    float4 s = *(const float4*)(src + (size_t)i * 4);
    float4 o = make_float4(d * s.x, d * s.y, d * s.z, d * s.w);
    *(float4*)(dst + (size_t)i * 4) = o;
  }
}

__global__ void k_self_init1(const float* __restrict__ src, const float* __restrict__ dis,
                             float* __restrict__ dst, int N) {
  int i = blockIdx.x * blockDim.x + threadIdx.x;
  if (i < N) {
    float d = dis[i];
    dst[i] = d * d * src[i];
  }
}

// dst[col,:] += dis[row]*dis[col] * src[row,:]
// One thread per (edge, 4-feature group): b128 gather + 4 L2-resident f32 atomics.
template<int F>
__global__ void k_edge_scatter4(const int* __restrict__ row, const int* __restrict__ colv,
                                const float* __restrict__ dis, const float* __restrict__ src,
                                float* __restrict__ dst, int E) {
  static_assert(F % 4 == 0, "F must be multiple of 4");
  constexpr int G = F / 4;
  unsigned int i = blockIdx.x * blockDim.x + threadIdx.x;
  if (i < (unsigned int)E * G) {
    unsigned int e = i / G;
    unsigned int g = i - e * G;
    int r = row[e], c = colv[e];
    float w = dis[r] * dis[c];
    const float4 s = *(const float4*)(src + (size_t)r * F + g * 4);
    float* d = dst + (size_t)c * F + g * 4;
    atomic_add_f32(d + 0, w * s.x);
    atomic_add_f32(d + 1, w * s.y);
    atomic_add_f32(d + 2, w * s.z);
    atomic_add_f32(d + 3, w * s.w);
  }
}

__global__ void k_edge_scatter1(const int* __restrict__ row, const int* __restrict__ colv,
                                const float* __restrict__ dis, const float* __restrict__ src,
                                float* __restrict__ dst, int E) {
  int e = blockIdx.x * blockDim.x + threadIdx.x;
  if (e < E) {
    int r = row[e], c = colv[e];
    atomic_add_f32(&dst[c], dis[r] * dis[c] * src[r]);
  }
}

// ---------------- dense GEMMs via V_WMMA_F32_16X16X4_F32 ----------------
// One wave computes a 16-row x 64-col strip: 4 accumulators, K=64 in steps of 4.
// A layout (16x4 f32): lanes 0-15 hold K=0,1 ; lanes 16-31 hold K=2,3 (M = lane%16).
// B/D layout: row striped across lanes within a VGPR (half-wave split like C/D).
__global__ void k_gcn2_gemm(const float* __restrict__ P, const float* __restrict__ X0,
                            const float* __restrict__ W, float* __restrict__ H, int N) {
  int wave = blockIdx.x * (blockDim.x >> 5) + (threadIdx.x >> 5);
  int lane = threadIdx.x & 31;
  int row0 = wave << 4;
  if (row0 >= N) return;              // wave-uniform exit: EXEC stays all-ones for WMMA
  int m  = lane & 15;
  int kh = lane >> 4;                 // which K-pair this half-wave holds
  int grow = row0 + m; if (grow > N - 1) grow = N - 1;
  size_t abase = (size_t)grow * 64;
  v8f acc[4] = {};
  for (int k0 = 0; k0 < 64; k0 += 4) {
    int ka = k0 + (kh << 1);
    v2f pv = *(const v2f*)(P + abase + ka);       // global_load_b64
    v2f xv = *(const v2f*)(X0 + abase + ka);      // global_load_b64
    v2f a = pv * (1.0f - GCN_ALPHA) + xv * GCN_ALPHA;
    const float* wb = W + (size_t)ka * 64 + m;    // rows ka, ka+1 of B-tile
#pragma unroll
    for (int nt = 0; nt < 4; ++nt) {
      v2f b;
      b.x = wb[nt * 16];
      b.y = wb[64 + nt * 16];
      acc[nt] = __builtin_amdgcn_wmma_f32_16x16x4_f32(
          false, a, false, b, (short)0, acc[nt], false, false);
    }
  }
  if (row0 + 16 <= N) {               // full tile: unconditional coalesced stores
#pragma unroll
    for (int v = 0; v < 8; ++v) {
      float* o = H + (size_t)(row0 + v + (kh << 3)) * 64 + m;
      o[0]  = fmaxf(acc[0][v], 0.f);
      o[16] = fmaxf(acc[1][v], 0.f);
      o[32] = fmaxf(acc[2][v], 0.f);
      o[48] = fmaxf(acc[3][v], 0.f);
    }
  } else {
#pragma unroll
    for (int v = 0; v < 8; ++v) {
      int r = row0 + v + (kh << 3);
      if (r < N) {
        float* o = H + (size_t)r * 64 + m;
        o[0]  = fmaxf(acc[0][v], 0.f);
        o[16] = fmaxf(acc[1][v], 0.f);
        o[32] = fmaxf(acc[2][v], 0.f);
        o[48] = fmaxf(acc[3][v], 0.f);
      }
    }
  }
}

// T1[N,16] = H[N,64] @ W1[64,16]   (no activation; bias applied after propagate)
__global__ void k_lin1(const float* __restrict__ Hs, const float* __restrict__ W1,
                       float* __restrict__ T1, int N) {
  int wave = blockIdx.x * (blockDim.x >> 5) + (threadIdx.x >> 5);
  int lane = threadIdx.x & 31;
  int row0 = wave << 4;
  if (row0 >= N) return;
  int m  = lane & 15;
  int kh = lane >> 4;
  int grow = row0 + m; if (grow > N - 1) grow = N - 1;
  size_t abase = (size_t)grow * 64;
  v8f acc = {};
  for (int k0 = 0; k0 < 64; k0 += 4) {
    int ka = k0 + (kh << 1);
    v2f a = *(const v2f*)(Hs + abase + ka);        // global_load_b64
    v2f b;
    b.x = W1[(size_t)ka * 16 + m];
    b.y = W1[(size_t)(ka + 1) * 16 + m];
    acc = __builtin_amdgcn_wmma_f32_16x16x4_f32(
        false, a, false, b, (short)0, acc, false, false);
  }
  if (row0 + 16 <= N) {
#pragma unroll
    for (int v = 0; v < 8; ++v)
      T1[(size_t)(row0 + v + (kh << 3)) * 16 + m] = acc[v];
  } else {
#pragma unroll
    for (int v = 0; v < 8; ++v) {
      int r = row0 + v + (kh << 3);
      if (r < N) T1[(size_t)r * 16 + m] = acc[v];
    }
  }
}

// t2[i] = sum_f (p1[i,f] + b1[f]) * w2[f]
__global__ void k_lin2(const float* __restrict__ p1, const float* __restrict__ b1,
                       const float* __restrict__ w2, float* __restrict__ t2, int N) {
  int i = blockIdx.x * blockDim.x + threadIdx.x;
  if (i < N) {
    float s = 0.f;
#pragma unroll
    for (int q = 0; q < 4; ++q) {
      float4 pv = *(const float4*)(p1 + (size_t)i * 16 + q * 4);
      float4 bv = *(const float4*)(b1 + q * 4);
      float4 wv = *(const float4*)(w2 + q * 4);
      s += (pv.x + bv.x) * wv.x + (pv.y + bv.y) * wv.y +
           (pv.z + bv.z) * wv.z + (pv.w + bv.w) * wv.w;
    }
    t2[i] = s;
  }
}

__global__ void k_final(const float* __restrict__ p2, const float* __restrict__ b2,
                        float* __restrict__ out, int N) {
  int i = blockIdx.x * blockDim.x + threadIdx.x;
  if (i < N) {
    float v = p2[i] + b2[0];
    out[i] = 1.f / (1.f + expf(-v));
  }
}

// ---------------- launch ----------------
extern "C" void kernel_launch(void* const* d_in, const int* in_sizes, int n_in,
                              void* d_out, int out_size, void* d_ws, size_t ws_size,
                              hipStream_t stream) {
  (void)n_in; (void)out_size; (void)ws_size;
  const float* x  = (const float*)d_in[0];
  const int*   ei = (const int*)d_in[1];
  const float* gw = (const float*)d_in[2];
  const float* w1 = (const float*)d_in[3];
  const float* b1 = (const float*)d_in[4];
  const float* w2 = (const float*)d_in[5];
  const float* b2 = (const float*)d_in[6];
  const int F = 64;
  const int N = in_sizes[0] / F;
  const int E = in_sizes[1] / 2;
  const int L = in_sizes[2] / (F * F);
  float* out = (float*)d_out;

  float* ws  = (float*)d_ws;
  float* dis = ws;                  ws += N;
  float* h   = ws;                  ws += (size_t)N * F;
  float* p   = ws;                  ws += (size_t)N * F;
  float* t1  = ws;                  ws += (size_t)N * 16;
  float* p1  = ws;                  ws += (size_t)N * 16;
  float* t2  = ws;                  ws += N;
  float* p2  = ws;

  const int* row = ei;
  const int* col = ei + E;
  const int B = 256;
  auto cdiv = [](long long a, long long b) { return (int)((a + b - 1) / b); };

  k_init_deg<<<cdiv(N, B), B, 0, stream>>>(dis, N);
  k_deg_accum<<<cdiv(E, B), B, 0, stream>>>(col, dis, E);
  k_make_dis<<<cdiv(N, B), B, 0, stream>>>(dis, N);

  const int gemm_blocks = cdiv(cdiv(N, 16), 8);  // 8 waves/block, 1 wave per 16-row strip
  for (int l = 0; l < L; ++l) {
    const float* src = (l == 0) ? x : h;
    k_self_init4<64><<<cdiv((long long)N * 16, B), B, 0, stream>>>(src, dis, p, N);
    k_edge_scatter4<64><<<cdiv((long long)E * 16, B), B, 0, stream>>>(row, col, dis, src, p, E);
    k_gcn2_gemm<<<gemm_blocks, B, 0, stream>>>(p, x, gw + (size_t)l * F * F, h, N);
  }

  k_lin1<<<gemm_blocks, B, 0, stream>>>(h, w1, t1, N);
  k_self_init4<16><<<cdiv((long long)N * 4, B), B, 0, stream>>>(t1, dis, p1, N);
  k_edge_scatter4<16><<<cdiv((long long)E * 4, B), B, 0, stream>>>(row, col, dis, t1, p1, E);

  k_lin2<<<cdiv(N, B), B, 0, stream>>>(p1, b1, w2, t2, N);
  k_self_init1<<<cdiv(N, B), B, 0, stream>>>(t2, dis, p2, N);
  k_edge_scatter1<<<cdiv(E, B), B, 0, stream>>>(row, col, dis, t2, p2, E);
  k_final<<<cdiv(N, B), B, 0, stream>>>(p2, b2, out, N);
}